// ParamSampler_77678778515631
// MI455X (gfx1250) — compile-verified
//
#include <hip/hip_runtime.h>
#include <hip/hip_bf16.h>
#include <stdint.h>

#define HW      512
#define HWHW    (HW * HW)
#define PAD     3
#define PROWS   (HW + 2 * PAD)   // 518
#define PCOLS   520              // padded pitch
#define MAXC    16384
#define NPROP   30
#define FEAT_C  256
#define TILE    16
#define TDIM    (TILE + 2 * PAD) // 22

#if defined(__has_builtin)
#  if __has_builtin(__builtin_amdgcn_tensor_load_to_lds) && \
      __has_builtin(__builtin_amdgcn_s_wait_tensorcnt)
#    define HAVE_TDM 1
#  endif
#endif
#ifndef HAVE_TDM
#  define HAVE_TDM 0
#endif

typedef unsigned int u32x4 __attribute__((ext_vector_type(4)));
typedef int          i32x4 __attribute__((ext_vector_type(4)));
typedef int          i32x8 __attribute__((ext_vector_type(8)));

// ---------------------------------------------------------------------------
// K1: center = mean of heatmap[:,17]; c2 = (center + avgpool3(center))/2,
//     into a -inf padded (PROWS x PCOLS) buffer (keeps NMS tile loads
//     in-range, required for async-to-LDS staging). 2D grid: no div/mod.
// ---------------------------------------------------------------------------
__global__ void k_center(const float* __restrict__ hm,
                         float* __restrict__ c2pad,
                         int* __restrict__ counter) {
    int px = blockIdx.x * blockDim.x + threadIdx.x;
    int py = blockIdx.y;
    if (px == 0 && py == 0) *counter = 0;     // reset candidate counter
    if (px >= PCOLS) return;
    int i = py * PCOLS + px;
    if (px < PAD || px >= PAD + HW || py < PAD || py >= PAD + HW) {
        c2pad[i] = -__builtin_inff();
        return;
    }
    int x = px - PAD, y = py - PAD;
    const float* h0 = hm + (size_t)17 * HWHW;  // batch 0, channel 17
    const float* h1 = hm + (size_t)35 * HWHW;  // batch 1, channel 17
    float s = 0.f, c = 0.f;
    #pragma unroll
    for (int dy = -1; dy <= 1; ++dy) {
        #pragma unroll
        for (int dx = -1; dx <= 1; ++dx) {
            int yy = y + dy, xx = x + dx;
            if ((unsigned)yy < HW && (unsigned)xx < HW) {   // pad contributes 0
                float v = 0.5f * (h0[yy * HW + xx] + h1[yy * HW + xx]);
                s += v;
                if (dy == 0 && dx == 0) c = v;
            }
        }
    }
    c2pad[i] = 0.5f * (c + s * (1.0f / 9.0f));  // count_include_pad: /9 always
}

// ---------------------------------------------------------------------------
// K2: 7x7 max-pool NMS. Stage 22x22 halo tile into LDS via the CDNA5 async
//     global->LDS path (ASYNCcnt), then compact surviving local maxima.
// ---------------------------------------------------------------------------
__global__ void k_nms(const float* __restrict__ c2pad,
                      float* __restrict__ cand_val,
                      int* __restrict__ cand_idx,
                      int* __restrict__ counter) {
    __shared__ float tile[TDIM * TDIM];      // 484 floats
    const int bx = blockIdx.x % (HW / TILE);
    const int by = blockIdx.x / (HW / TILE);
    const int tid = threadIdx.x;
    const int lx = tid % TILE, ly = tid / TILE;
    const int r0 = by * TILE, c0 = bx * TILE;   // padded-space tile origin

    // async per-lane gather into LDS: each lane copies one global dword to its LDS slot
    for (int e = tid; e < TDIM * TDIM; e += TILE * TILE) {
        int tr = e / TDIM, tc = e % TDIM;
        const float* g = c2pad + (size_t)(r0 + tr) * PCOLS + (c0 + tc);
        uint32_t lds_addr = (uint32_t)(uintptr_t)(&tile[e]);
        asm volatile("global_load_async_to_lds_b32 %0, %1, off"
                     :: "v"(lds_addr), "v"((unsigned long long)(uintptr_t)g)
                     : "memory");
    }
    asm volatile("s_wait_asynccnt 0" ::: "memory");
    __syncthreads();

    float c = tile[(ly + PAD) * TDIM + (lx + PAD)];
    float m = -__builtin_inff();
    #pragma unroll
    for (int dy = 0; dy < 7; ++dy)
        #pragma unroll
        for (int dx = 0; dx < 7; ++dx)
            m = fmaxf(m, tile[(ly + dy) * TDIM + (lx + dx)]);

    // strict window max survives; scores <= 0.01 can never reach the
    // returned top-30 (all top-30 exceed KEYPOINT_THRE), so pre-filter.
    if (m == c && c > 0.01f) {
        int slot = atomicAdd(counter, 1);
        if (slot < MAXC) {
            cand_val[slot] = c;
            cand_idx[slot] = (r0 + ly) * HW + (c0 + lx);
        }
    }
}

// ---------------------------------------------------------------------------
// K3: single-block exact top-30 with jax tie-break (value desc, index asc).
// ---------------------------------------------------------------------------
__global__ void k_topk(float* __restrict__ cand_val,
                       const int* __restrict__ cand_idx,
                       const int* __restrict__ counter,
                       float* __restrict__ out,
                       int* __restrict__ top_idx) {
    __shared__ float rv[256];
    __shared__ int   ri[256];
    __shared__ int   rs[256];
    __shared__ float s_score[NPROP];
    __shared__ int   s_idx[NPROP];
    const int tid = threadIdx.x;
    int nc = *counter;
    if (nc > MAXC) nc = MAXC;

    for (int round = 0; round < NPROP; ++round) {
        float bv = -__builtin_inff();
        int bi = 0x7fffffff, bs = -1;
        for (int j = tid; j < nc; j += 256) {
            float v = cand_val[j];
            int id = cand_idx[j];
            if (v > bv || (v == bv && id < bi)) { bv = v; bi = id; bs = j; }
        }
        rv[tid] = bv; ri[tid] = bi; rs[tid] = bs;
        __syncthreads();
        for (int off = 128; off > 0; off >>= 1) {
            if (tid < off) {
                float v2 = rv[tid + off]; int i2 = ri[tid + off];
                if (v2 > rv[tid] || (v2 == rv[tid] && i2 < ri[tid])) {
                    rv[tid] = v2; ri[tid] = i2; rs[tid] = rs[tid + off];
                }
            }
            __syncthreads();
        }
        if (tid == 0) {
            s_score[round] = rv[0];
            s_idx[round]   = ri[0];
            if (rs[0] >= 0) cand_val[rs[0]] = -__builtin_inff();  // retire winner
        }
        __syncthreads();
    }

    if (tid < NPROP) {
        int id = s_idx[tid];
        float sc = s_score[tid];
        int y = id / HW, x = id % HW;
        // out layout: coord2[60,2] | imgid[60] | param[60,256] | score[60]
        out[2 * tid + 0] = (float)y;
        out[2 * tid + 1] = (float)x;
        out[2 * (tid + NPROP) + 0] = (float)y;
        out[2 * (tid + NPROP) + 1] = (float)x;
        out[120 + tid]          = 0.0f;
        out[120 + NPROP + tid]  = 1.0f;
        out[15540 + tid]         = sc;
        out[15540 + NPROP + tid] = sc;
        top_idx[tid] = id;
    }
}

// ---------------------------------------------------------------------------
// K4: instance_param gather. Row j = features[j/30][:, y, x]: 256 floats
//     strided 1 MB apart. Use the Tensor Data Mover: one D# describing a
//     (tile_dim0=1 x tile_dim1=256) tile with tensor_dim0_stride = HW*HW
//     DMAs the whole channel column into contiguous LDS (TENSORcnt path).
// ---------------------------------------------------------------------------
__global__ void k_gather(const float* __restrict__ feat,
                         const int* __restrict__ top_idx,
                         float* __restrict__ out) {
    const int j = blockIdx.x;            // 0..59
    const int tid = threadIdx.x;         // 0..255
    const int b = j / NPROP, n = j % NPROP;
    const int id = top_idx[n];
    const int y = id / HW, x = id % HW;

#if HAVE_TDM
    __shared__ float buf[FEAT_C];
    if (tid < 32) {                      // single wave issues the DMA
        uint64_t gaddr = (uint64_t)(uintptr_t)
            (feat + (size_t)b * FEAT_C * HWHW + (size_t)y * HW + x);
        uint32_t laddr = (uint32_t)(uintptr_t)(&buf[0]);
        // D# group 0: count=1 | lds_addr | global_addr | type=2
        u32x4 g0 = { 1u,
                     laddr,
                     (uint32_t)gaddr,
                     (uint32_t)((gaddr >> 32) & 0x01FFFFFFu) | (2u << 30) };
        // D# group 1: data_size=4B; tensor_dim0=HW, tensor_dim1=FEAT_C;
        //             tile_dim0=1, tile_dim1=FEAT_C; dim0_stride=HW*HW
        i32x8 g1 = { (int)(2u << 16),                       // wg_mask=0, data_size=4B
                     (int)((uint32_t)HW << 16),             // tensor_dim0[15:0]
                     (int)((uint32_t)FEAT_C << 16),         // tensor_dim1[15:0]
                     (int)(1u << 16),                       // tile_dim0=1
                     (int)FEAT_C,                           // tile_dim1=256
                     (int)HWHW,                             // tensor_dim0_stride lo32
                     0, 0 };
        i32x4 gz = { 0, 0, 0, 0 };
#  if __clang_major__ >= 23
        i32x8 gz8 = { 0, 0, 0, 0, 0, 0, 0, 0 };
        __builtin_amdgcn_tensor_load_to_lds(g0, g1, gz, gz, gz8, 0);
#  else
        __builtin_amdgcn_tensor_load_to_lds(g0, g1, gz, gz, 0);
#  endif
        __builtin_amdgcn_s_wait_tensorcnt(0);
    }
    __syncthreads();
    out[180 + (size_t)j * FEAT_C + tid] = buf[tid];
#else
    size_t off = (((size_t)(b * FEAT_C + tid)) * HW + y) * HW + x;
    out[180 + (size_t)j * FEAT_C + tid] = feat[off];
#endif
}

// ---------------------------------------------------------------------------
extern "C" void kernel_launch(void* const* d_in, const int* in_sizes, int n_in,
                              void* d_out, int out_size, void* d_ws, size_t ws_size,
                              hipStream_t stream) {
    const float* feat = (const float*)d_in[0];  // (2,256,512,512)
    const float* hm   = (const float*)d_in[1];  // (2,18,512,512)
    float* out = (float*)d_out;                 // 15600 floats

    char* ws = (char*)d_ws;
    size_t off_c2   = 0;
    size_t off_cnt  = off_c2  + (size_t)PROWS * PCOLS * sizeof(float);
    size_t off_cidx = off_cnt + 16;
    size_t off_cval = off_cidx + (size_t)MAXC * sizeof(int);
    size_t off_tidx = off_cval + (size_t)MAXC * sizeof(float);

    float* c2pad    = (float*)(ws + off_c2);
    int*   counter  = (int*)  (ws + off_cnt);
    int*   cand_idx = (int*)  (ws + off_cidx);
    float* cand_val = (float*)(ws + off_cval);
    int*   top_idx  = (int*)  (ws + off_tidx);

    hipLaunchKernelGGL(k_center, dim3((PCOLS + 255) / 256, PROWS), dim3(256), 0, stream,
                       hm, c2pad, counter);
    hipLaunchKernelGGL(k_nms, dim3((HW / TILE) * (HW / TILE)), dim3(TILE * TILE), 0, stream,
                       c2pad, cand_val, cand_idx, counter);
    hipLaunchKernelGGL(k_topk, dim3(1), dim3(256), 0, stream,
                       cand_val, cand_idx, counter, out, top_idx);
    hipLaunchKernelGGL(k_gather, dim3(2 * NPROP), dim3(FEAT_C), 0, stream,
                       feat, top_idx, out);
}